// GNN_DDI_30932354466097
// MI455X (gfx1250) — compile-verified
//
#include <hip/hip_runtime.h>
#include <math.h>

#define EMB 64
#define NEG_SLOPE 0.2f

typedef __attribute__((ext_vector_type(2))) float v2f;
typedef __attribute__((ext_vector_type(8))) float v8f;

// ---------------------------------------------------------------------------
// float atomic max via signed/unsigned integer atomics (valid for all
// non-NaN floats; nmax initialized to -inf bit pattern).
// ---------------------------------------------------------------------------
__device__ __forceinline__ void atomicMaxFloat(float* addr, float val) {
    if (val >= 0.0f) {
        atomicMax((int*)addr, __float_as_int(val));
    } else {
        atomicMin((unsigned int*)addr, __float_as_uint(val));
    }
}

// ---------------------------------------------------------------------------
// z = h @ W  using V_WMMA_F32_16X16X4_F32 (full fp32 precision).
// One wave per 16-row slab; 4 column tiles of 16 -> 16x64 output per wave.
// W transposed into LDS: Wt[n*64 + k] = W[k*64 + n].
// A frag (16x4 f32): lanes 0-15 carry K=k,k+1 ; lanes 16-31 carry K=k+2,k+3.
// C/D (16x16 f32): VGPR v, lane L -> M = v + 8*(L>=16), N = L%16.
// ---------------------------------------------------------------------------
__global__ void __launch_bounds__(256)
gat_gemm_xw(const float* __restrict__ h, const float* __restrict__ W,
            float* __restrict__ z, int n) {
    __shared__ float Wt[EMB * EMB];
    for (int i = 0; i < 16; ++i) {
        int lin = i * 256 + threadIdx.x;      // coalesced read of W
        int k = lin >> 6, nn = lin & 63;
        Wt[nn * EMB + k] = W[lin];
    }
    __syncthreads();

    const int wave = threadIdx.x >> 5;
    const int lane = threadIdx.x & 31;
    const int slab = blockIdx.x * 8 + wave;
    const int rowBase = slab * 16;
    if (rowBase >= n) return;

    const int halfSel = lane >> 4;            // 0: lanes 0-15, 1: lanes 16-31
    const int lm = lane & 15;
    const float* hrow = h + (size_t)(rowBase + lm) * EMB;

    v8f c[4];
    for (int t = 0; t < 4; ++t) c[t] = (v8f){0.f, 0.f, 0.f, 0.f, 0.f, 0.f, 0.f, 0.f};

    for (int k = 0; k < EMB; k += 4) {
        const int kk = k + 2 * halfSel;
        v2f a = *(const v2f*)(hrow + kk);
#pragma unroll
        for (int t = 0; t < 4; ++t) {
            v2f b = *(const v2f*)(&Wt[(t * 16 + lm) * EMB + kk]);
            c[t] = __builtin_amdgcn_wmma_f32_16x16x4_f32(
                false, a, false, b, (short)0, c[t], false, false);
        }
    }

#pragma unroll
    for (int t = 0; t < 4; ++t)
#pragma unroll
        for (int v = 0; v < 8; ++v)
            z[(size_t)(rowBase + v + 8 * halfSel) * EMB + t * 16 + lm] = c[t][v];
}

// ---------------------------------------------------------------------------
// Per-node attention scores: ssrc[n] = z[n].a_src, sdst[n] = z[n].a_dst.
// One wave per node; float2 per lane; butterfly reduce (wave32).
// ---------------------------------------------------------------------------
__global__ void __launch_bounds__(256)
gat_node_scores(const float* __restrict__ z, const float* __restrict__ asrc,
                const float* __restrict__ adst, float* __restrict__ ssrc,
                float* __restrict__ sdst, int n) {
    const int wave = threadIdx.x >> 5;
    const int lane = threadIdx.x & 31;
    const int node = blockIdx.x * 8 + wave;
    if (node >= n) return;
    v2f zv = *(const v2f*)(z + (size_t)node * EMB + 2 * lane);
    v2f av = *(const v2f*)(asrc + 2 * lane);
    v2f dv = *(const v2f*)(adst + 2 * lane);
    float s1 = zv.x * av.x + zv.y * av.y;
    float s2 = zv.x * dv.x + zv.y * dv.y;
#pragma unroll
    for (int off = 16; off > 0; off >>= 1) {
        s1 += __shfl_xor(s1, off, 32);
        s2 += __shfl_xor(s2, off, 32);
    }
    if (lane == 0) { ssrc[node] = s1; sdst[node] = s2; }
}

// acc = 0 ; nmax = -inf ; nsum = 0
__global__ void gat_init(float* __restrict__ acc, float* __restrict__ nmax,
                         float* __restrict__ nsum, int n) {
    int i = blockIdx.x * blockDim.x + threadIdx.x;
    if (i < n * EMB) acc[i] = 0.0f;
    if (i < n) { nmax[i] = -INFINITY; nsum[i] = 0.0f; }
}

// e = leaky_relu(ssrc[src] + sdst[dst]) ; segment max over dst
__global__ void gat_edge_max(const int* __restrict__ src, const int* __restrict__ dst,
                             const float* __restrict__ ssrc, const float* __restrict__ sdst,
                             float* __restrict__ ebuf, float* __restrict__ nmax,
                             int E, int EP) {
    int i = blockIdx.x * blockDim.x + threadIdx.x;
    if (i >= EP) return;
    int s, d;
    if (i < E) { s = src[i]; d = dst[i]; } else { s = d = i - E; }  // self loops
    float e = ssrc[s] + sdst[d];
    e = (e > 0.0f) ? e : NEG_SLOPE * e;
    ebuf[i] = e;
    atomicMaxFloat(&nmax[d], e);
}

// ex = exp(e - max[dst]) ; segment sum over dst
__global__ void gat_edge_expsum(const int* __restrict__ dst, float* __restrict__ ebuf,
                                const float* __restrict__ nmax, float* __restrict__ nsum,
                                int E, int EP) {
    int i = blockIdx.x * blockDim.x + threadIdx.x;
    if (i >= EP) return;
    int d = (i < E) ? dst[i] : (i - E);
    float ex = __expf(ebuf[i] - nmax[d]);
    ebuf[i] = ex;
    atomicAdd(&nsum[d], ex);
}

// acc[dst] += (ex / (sum[dst]+eps)) * z[src] ; one wave per edge, float2/lane
__global__ void __launch_bounds__(256)
gat_edge_scatter(const int* __restrict__ src, const int* __restrict__ dst,
                 const float* __restrict__ ebuf, const float* __restrict__ nsum,
                 const float* __restrict__ z, float* __restrict__ acc,
                 int E, int EP) {
    const int gw = (int)((blockIdx.x * blockDim.x + threadIdx.x) >> 5);
    const int lane = threadIdx.x & 31;
    if (gw >= EP) return;
    int s, d;
    if (gw < E) { s = src[gw]; d = dst[gw]; } else { s = d = gw - E; }
    float alpha = ebuf[gw] / (nsum[d] + 1e-16f);
    v2f zv = *(const v2f*)(z + (size_t)s * EMB + 2 * lane);
    float* out = acc + (size_t)d * EMB + 2 * lane;
    atomicAdd(out, alpha * zv.x);
    atomicAdd(out + 1, alpha * zv.y);
}

// acc = acc + b (+ relu for non-final layers), in place
__global__ void gat_finalize(float* __restrict__ acc, const float* __restrict__ b,
                             int n, int do_relu) {
    int i = blockIdx.x * blockDim.x + threadIdx.x;
    if (i >= n * EMB) return;
    float v = acc[i] + b[i & 63];
    if (do_relu) v = fmaxf(v, 0.0f);
    acc[i] = v;
}

extern "C" void kernel_launch(void* const* d_in, const int* in_sizes, int n_in,
                              void* d_out, int out_size, void* d_ws, size_t ws_size,
                              hipStream_t stream) {
    const float* x    = (const float*)d_in[0];   // [N,64]
    const int*   ei   = (const int*)d_in[1];     // [2,E] (int32; see analysis)
    const float* Ws   = (const float*)d_in[2];   // [3,64,64]
    const float* asrc = (const float*)d_in[3];   // [3,64]
    const float* adst = (const float*)d_in[4];   // [3,64]
    const float* bs   = (const float*)d_in[5];   // [3,64]

    const int n  = in_sizes[0] / EMB;
    const int E  = in_sizes[1] / 2;
    const int EP = E + n;                        // edges + self loops
    const int* src = ei;
    const int* dst = ei + E;

    float* ws   = (float*)d_ws;
    float* hA   = ws;
    float* hB   = hA + (size_t)n * EMB;
    float* z    = hB + (size_t)n * EMB;
    float* ssrc = z + (size_t)n * EMB;
    float* sdst = ssrc + n;
    float* nmax = sdst + n;
    float* nsum = nmax + n;
    float* ebuf = nsum + n;                      // EP floats

    const int slabBlocks = ((n + 15) / 16 + 7) / 8;
    const int nodeBlocks = (n + 7) / 8;
    const int nvBlocks   = (n * EMB + 255) / 256;
    const int epBlocks   = (EP + 255) / 256;
    const int scatBlocks = (EP + 7) / 8;         // 8 waves/block, wave per edge

    const float* hin = x;
    for (int l = 0; l < 3; ++l) {
        float* acc = (l == 0) ? hA : (l == 1) ? hB : (float*)d_out;
        const float* W  = Ws   + l * EMB * EMB;
        const float* as = asrc + l * EMB;
        const float* ad = adst + l * EMB;
        const float* b  = bs   + l * EMB;

        gat_gemm_xw     <<<slabBlocks, 256, 0, stream>>>(hin, W, z, n);
        gat_node_scores <<<nodeBlocks, 256, 0, stream>>>(z, as, ad, ssrc, sdst, n);
        gat_init        <<<nvBlocks,   256, 0, stream>>>(acc, nmax, nsum, n);
        gat_edge_max    <<<epBlocks,   256, 0, stream>>>(src, dst, ssrc, sdst, ebuf, nmax, E, EP);
        gat_edge_expsum <<<epBlocks,   256, 0, stream>>>(dst, ebuf, nmax, nsum, E, EP);
        gat_edge_scatter<<<scatBlocks, 256, 0, stream>>>(src, dst, ebuf, nsum, z, acc, E, EP);
        gat_finalize    <<<nvBlocks,   256, 0, stream>>>(acc, b, n, l != 2);

        hin = acc;
    }
}